// GAT_21947282882668
// MI455X (gfx1250) — compile-verified
//
#include <hip/hip_runtime.h>

// ---------------------------------------------------------------------------
// GAT pipeline for MI455X (gfx1250, wave32).
// Dense GEMMs: V_WMMA_F32_16X16X4_F32, A-panel staged to LDS via the Tensor
// Data Mover (tensor_load_to_lds + s_wait_tensorcnt), B streamed from L2.
// Attention softmax/scatter via atomics (HBM-bandwidth bound phase).
// ---------------------------------------------------------------------------

typedef __attribute__((ext_vector_type(2))) float    v2f;
typedef __attribute__((ext_vector_type(8))) float    v8f;
typedef __attribute__((ext_vector_type(4))) unsigned v4u;
typedef __attribute__((ext_vector_type(8))) int      v8i_;
typedef __attribute__((ext_vector_type(4))) int      v4i_;

#define SELU_SCALE 1.0507009873554805f
#define SELU_ALPHA 1.6732632423543772f

__device__ __forceinline__ float selu_f(float x) {
  return SELU_SCALE * (x > 0.f ? x : SELU_ALPHA * (__expf(x) - 1.f));
}

// Order-preserving float <-> uint encoding for atomicMax-based segment max.
__device__ __forceinline__ unsigned fenc(float f) {
  unsigned b = __float_as_uint(f);
  return b ^ ((b & 0x80000000u) ? 0xFFFFFFFFu : 0x80000000u);
}
__device__ __forceinline__ float fdec(unsigned k) {
  unsigned b = k ^ ((k & 0x80000000u) ? 0x80000000u : 0xFFFFFFFFu);
  return __uint_as_float(b);
}

// ---------------------------------------------------------------------------
// TDM: DMA the contiguous 16xK fp32 A-panel from global into LDS.
// D# per CDNA5 ISA 8.3/8.4:
//   group0: count=1 | lds_addr[63:32] | global_addr[120:64] | type=2[127:126]
//   group1: data_size=4B[17:16], tensor_dim0=K[79:48], tensor_dim1=16[111:80],
//           tile_dim0=K[127:112], tile_dim1=16[143:128],
//           tensor_dim0_stride=K[207:160]
// ---------------------------------------------------------------------------
#if defined(__has_builtin)
#if __has_builtin(__builtin_amdgcn_tensor_load_to_lds)
#define HAVE_TDM 1
#endif
#endif

__device__ __forceinline__ void tdm_load_A(const float* gA, float* lds, int K) {
#ifdef HAVE_TDM
  unsigned lds_off = (unsigned)(unsigned long long)(void*)lds;  // addr[31:0] = LDS offset
  unsigned long long ga = (unsigned long long)gA;
  v4u g0;
  g0.x = 1u;                                                   // count=1, user D#
  g0.y = lds_off;                                              // lds_addr
  g0.z = (unsigned)(ga & 0xFFFFFFFFull);                       // global_addr[31:0]
  g0.w = (unsigned)((ga >> 32) & 0x1FFFFFFull) | (2u << 30);   // addr[56:32] | type=2
  v8i_ g1;
  g1[0] = 0x00020000;                      // workgroup_mask=0, data_size=2 (4B)
  g1[1] = (int)((unsigned)(K & 0xFFFF) << 16);  // tensor_dim0[15:0] (rows are K elems)
  g1[2] = (int)(16u << 16);                // tensor_dim0 hi=0 | tensor_dim1[15:0]=16
  g1[3] = (int)((unsigned)(K & 0xFFFF) << 16);  // tensor_dim1 hi=0 | tile_dim0=K
  g1[4] = 16;                              // tile_dim1=16, tile_dim2=0
  g1[5] = K;                               // tensor_dim0_stride[31:0]=K
  g1[6] = 0;                               // stride hi, dim1_stride lo
  g1[7] = 0;
  v4i_ z4 = {};
#if __clang_major__ >= 23
  v8i_ z8 = {};
  __builtin_amdgcn_tensor_load_to_lds(g0, g1, z4, z4, z8, 0);
#else
  __builtin_amdgcn_tensor_load_to_lds(g0, g1, z4, z4, 0);
#endif
  __builtin_amdgcn_s_wait_tensorcnt(0);
#else
  // Fallback: cooperative coalesced copy (panel is contiguous).
  for (int i = threadIdx.x & 31; i < 16 * K; i += 32) lds[i] = gA[i];
#endif
}

// ---------------------------------------------------------------------------
// WMMA fp32 GEMM:  C[M,Nc] = act(A[M,K] @ B[K,Nc] + bias)
// Requires M%16==0, Nc%32==0, K%4==0 (true for every GEMM in this model).
// Block = 256 threads = 8 waves. The block's 16xK A-panel lives in LDS
// (TDM-loaded once); wave w computes a 16x32 tile at col (blockIdx.x*8+w)*32,
// reusing each A fragment for two WMMAs. Early-outs are wave-uniform so EXEC
// is all-ones at every v_wmma.
//
// Fragment layout per CDNA5 ISA 7.12.2:
//   A 16x4 fp32:  lanes 0-15 -> M=lane, K=k+0/k+1; lanes 16-31 -> K=k+2/k+3.
//   B 4x16 fp32:  mirrored (row striped across lanes within a VGPR).
//   C/D 16x16:    VGPR r: lanes 0-15 -> (M=r, N=lane); lanes 16-31 -> (M=r+8).
// ---------------------------------------------------------------------------
__global__ __launch_bounds__(256) void k_wmma_gemm(
    const float* __restrict__ A, const float* __restrict__ B,
    const float* __restrict__ bias, float* __restrict__ C,
    int M, int K, int Nc, int act) {
  extern __shared__ float ldsA[];  // [16][K], row-major (== contiguous panel)
  const int tid  = threadIdx.x;
  const int lane = tid & 31;
  const int wave = tid >> 5;
  const int row0 = blockIdx.y << 4;
  const int col0 = (blockIdx.x * 8 + wave) << 5;  // 32 cols per wave

  if (wave == 0) tdm_load_A(A + (size_t)row0 * K, ldsA, K);
  __syncthreads();

  if (col0 >= Nc) return;  // wave-uniform

  const int half = lane >> 4;  // 0: lanes 0-15, 1: lanes 16-31
  const int l16  = lane & 15;

  const float* Alds = ldsA + l16 * K + half * 2;
  const float* Bp0  = B + (size_t)(half * 2) * Nc + col0 + l16;
  const float* Bp1  = Bp0 + 16;

  v8f acc0 = {}, acc1 = {};
  for (int k = 0; k < K; k += 4) {
    __builtin_prefetch(Bp0 + (size_t)(k + 8) * Nc, 0, 1);  // global_prefetch_b8
    v2f a, b0, b1;
    a.x = Alds[k];
    a.y = Alds[k + 1];
    size_t ro = (size_t)k * Nc;
    b0.x = Bp0[ro];
    b0.y = Bp0[ro + Nc];
    b1.x = Bp1[ro];
    b1.y = Bp1[ro + Nc];
    acc0 = __builtin_amdgcn_wmma_f32_16x16x4_f32(false, a, false, b0, (short)0,
                                                 acc0, false, false);
    acc1 = __builtin_amdgcn_wmma_f32_16x16x4_f32(false, a, false, b1, (short)0,
                                                 acc1, false, false);
  }

  const float bv0 = bias ? bias[col0 + l16] : 0.f;
  const float bv1 = bias ? bias[col0 + 16 + l16] : 0.f;
  float* Cp = C + (size_t)(row0 + half * 8) * Nc + col0 + l16;
#pragma unroll
  for (int r = 0; r < 8; ++r) {
    float v0 = acc0[r] + bv0;
    float v1 = acc1[r] + bv1;
    if (act) { v0 = selu_f(v0); v1 = selu_f(v1); }
    Cp[(size_t)r * Nc]      = v0;
    Cp[(size_t)r * Nc + 16] = v1;
  }
}

// ---------------------------------------------------------------------------
// Attention helpers (H=4 heads, C=64 channels, HC=256 fixed).
// ---------------------------------------------------------------------------

// Per-node attention logits: as[n,h] = sum_c h[n,h,c]*a_src[h,c] (same for ad)
__global__ void k_att_scores(const float* __restrict__ h,
                             const float* __restrict__ a_s,
                             const float* __restrict__ a_d,
                             float* __restrict__ as_, float* __restrict__ ad_,
                             int N) {
  int t = blockIdx.x * blockDim.x + threadIdx.x;
  if (t >= N * 4) return;
  int n = t >> 2, hd = t & 3;
  const float* hp = h + (size_t)n * 256 + hd * 64;
  const float* sp = a_s + hd * 64;
  const float* dp = a_d + hd * 64;
  float ss = 0.f, sd = 0.f;
#pragma unroll 8
  for (int c = 0; c < 64; ++c) {
    float v = hp[c];
    ss += v * sp[c];
    sd += v * dp[c];
  }
  as_[t] = ss;
  ad_[t] = sd;
}

// Edge pass 1: leaky-relu logits + per-dst running max (encoded atomicMax).
__global__ void k_edge_logits(const int* __restrict__ src,
                              const int* __restrict__ dst,
                              const float* __restrict__ as_,
                              const float* __restrict__ ad_,
                              float* __restrict__ elog, unsigned* __restrict__ mx,
                              int E, int Et) {
  int et = blockIdx.x * blockDim.x + threadIdx.x;
  if (et >= Et) return;
  int s, d;
  if (et < E) { s = src[et]; d = dst[et]; } else { s = d = et - E; }  // self-loop
#pragma unroll
  for (int h = 0; h < 4; ++h) {
    float e = as_[s * 4 + h] + ad_[d * 4 + h];
    e = e > 0.f ? e : 0.2f * e;  // leaky_relu, slope 0.2
    elog[(size_t)et * 4 + h] = e;
    atomicMax(&mx[d * 4 + h], fenc(e));
  }
}

// Edge pass 2: ex = exp(e - max[dst]); per-dst sum.
__global__ void k_edge_expsum(const int* __restrict__ dst,
                              float* __restrict__ elog,
                              const unsigned* __restrict__ mx,
                              float* __restrict__ sm, int E, int Et) {
  int et = blockIdx.x * blockDim.x + threadIdx.x;
  if (et >= Et) return;
  int d = (et < E) ? dst[et] : (et - E);
#pragma unroll
  for (int h = 0; h < 4; ++h) {
    float m = fdec(mx[d * 4 + h]);
    float v = __expf(elog[(size_t)et * 4 + h] - m);
    elog[(size_t)et * 4 + h] = v;
    atomicAdd(&sm[d * 4 + h], v);
  }
}

// Edge pass 3: out[dst] += h[src] * alpha.  One block per edge, 256 channels.
__global__ __launch_bounds__(256) void k_edge_agg(
    const int* __restrict__ src, const int* __restrict__ dst,
    const float* __restrict__ h, const float* __restrict__ elog,
    const float* __restrict__ sm, float* __restrict__ out, int E) {
  int et = blockIdx.x;
  int c = threadIdx.x;
  int s, d;
  if (et < E) { s = src[et]; d = dst[et]; } else { s = d = et - E; }
  int hd = c >> 6;
  float alpha = elog[(size_t)et * 4 + hd] / (sm[d * 4 + hd] + 1e-16f);
  atomicAdd(&out[(size_t)d * 256 + c], h[(size_t)s * 256 + c] * alpha);
}

// ---------------------------------------------------------------------------
// Pooling: concat(mean, max) per graph over [N,64] node features.
// ---------------------------------------------------------------------------
__global__ void k_pool_acc(const float* __restrict__ x,
                           const int* __restrict__ batch,
                           float* __restrict__ psum, unsigned* __restrict__ pmax,
                           float* __restrict__ cnt, int N) {
  int t = blockIdx.x * blockDim.x + threadIdx.x;
  if (t >= N * 64) return;
  int n = t >> 6, c = t & 63;
  int g = batch[n];
  float v = x[t];
  atomicAdd(&psum[g * 64 + c], v);
  atomicMax(&pmax[g * 64 + c], fenc(v));
  if (c == 0) atomicAdd(&cnt[g], 1.0f);
}

__global__ void k_pool_fin(const float* __restrict__ psum,
                           const unsigned* __restrict__ pmax,
                           const float* __restrict__ cnt,
                           float* __restrict__ cat, int colOff, int G) {
  int t = blockIdx.x * blockDim.x + threadIdx.x;
  if (t >= G * 64) return;
  int g = t >> 6, c = t & 63;
  float n = fmaxf(cnt[g], 1.0f);
  cat[g * 256 + colOff + c] = psum[t] / n;
  cat[g * 256 + colOff + 64 + c] = fdec(pmax[t]);
}

// ---------------------------------------------------------------------------
// Fills / init
// ---------------------------------------------------------------------------
__global__ void k_fill_u32(unsigned* p, unsigned v, size_t n) {
  size_t t = (size_t)blockIdx.x * blockDim.x + threadIdx.x;
  if (t < n) p[t] = v;
}
__global__ void k_init_bias(float* out, const float* __restrict__ bias, size_t n) {
  size_t t = (size_t)blockIdx.x * blockDim.x + threadIdx.x;
  if (t < n) out[t] = bias[t & 255];  // Nc == 256
}

// ---------------------------------------------------------------------------
// Host orchestration
// ---------------------------------------------------------------------------
static inline unsigned cdivu(size_t a, size_t b) { return (unsigned)((a + b - 1) / b); }

static void gemm(const float* A, const float* B, const float* bias, float* C,
                 int M, int K, int Nc, int act, hipStream_t st) {
  dim3 g((Nc + 255) / 256, M / 16);
  size_t shmem = (size_t)16 * K * sizeof(float);
  k_wmma_gemm<<<g, 256, shmem, st>>>(A, B, bias, C, M, K, Nc, act);
}

static void run_attention(const float* h, const float* a_s, const float* a_d,
                          const float* bvec, const int* src, const int* dst,
                          float* as_, float* ad_, unsigned* mx, float* sm,
                          float* elog, float* out, int N, int E, int Et,
                          hipStream_t st) {
  const int B = 256;
  k_att_scores<<<cdivu((size_t)N * 4, B), B, 0, st>>>(h, a_s, a_d, as_, ad_, N);
  k_fill_u32<<<cdivu((size_t)N * 4, B), B, 0, st>>>(mx, 0u, (size_t)N * 4);
  k_fill_u32<<<cdivu((size_t)N * 4, B), B, 0, st>>>((unsigned*)sm, 0u, (size_t)N * 4);
  k_edge_logits<<<cdivu(Et, B), B, 0, st>>>(src, dst, as_, ad_, elog, mx, E, Et);
  k_edge_expsum<<<cdivu(Et, B), B, 0, st>>>(dst, elog, mx, sm, E, Et);
  k_init_bias<<<cdivu((size_t)N * 256, B), B, 0, st>>>(out, bvec, (size_t)N * 256);
  k_edge_agg<<<Et, 256, 0, st>>>(src, dst, h, elog, sm, out, E);
}

static void run_pool(const float* xnode, const int* batch, float* psum,
                     unsigned* pmax, float* cnt, float* cat, int colOff,
                     int N, int G, hipStream_t st) {
  const int B = 256;
  k_fill_u32<<<cdivu((size_t)G * 64, B), B, 0, st>>>((unsigned*)psum, 0u, (size_t)G * 64);
  k_fill_u32<<<cdivu((size_t)G * 64, B), B, 0, st>>>(pmax, 0u, (size_t)G * 64);
  k_fill_u32<<<cdivu((size_t)G, B), B, 0, st>>>((unsigned*)cnt, 0u, (size_t)G);
  k_pool_acc<<<cdivu((size_t)N * 64, B), B, 0, st>>>(xnode, batch, psum, pmax, cnt, N);
  k_pool_fin<<<cdivu((size_t)G * 64, B), B, 0, st>>>(psum, pmax, cnt, cat, colOff, G);
}

extern "C" void kernel_launch(void* const* d_in, const int* in_sizes, int n_in,
                              void* d_out, int out_size, void* d_ws, size_t ws_size,
                              hipStream_t stream) {
  // ---- inputs (setup_inputs order) ----
  const float* x     = (const float*)d_in[0];
  const int*   ei    = (const int*)d_in[1];
  const int*   batch = (const int*)d_in[2];
  const float* g1w = (const float*)d_in[3],  *g1as = (const float*)d_in[4];
  const float* g1ad = (const float*)d_in[5], *g1b = (const float*)d_in[6];
  const float* l1w = (const float*)d_in[7],  *l1b = (const float*)d_in[8];
  const float* g2w = (const float*)d_in[9],  *g2as = (const float*)d_in[10];
  const float* g2ad = (const float*)d_in[11],*g2b = (const float*)d_in[12];
  const float* l2w = (const float*)d_in[13], *l2b = (const float*)d_in[14];
  const float* smw = (const float*)d_in[15], *smb = (const float*)d_in[16];
  const float* s1w = (const float*)d_in[17], *s1b = (const float*)d_in[18];
  const float* s2w = (const float*)d_in[19], *s2b = (const float*)d_in[20];
  const float* s3w = (const float*)d_in[21], *s3b = (const float*)d_in[22];
  const float* rw  = (const float*)d_in[23], *rb = (const float*)d_in[24];
  float* out = (float*)d_out;

  const int F_in = in_sizes[3] / 256;   // 260
  const int N    = in_sizes[0] / F_in;  // 100000
  const int E    = in_sizes[1] / 2;     // 800000
  const int Et   = E + N;               // + self loops
  const int G    = 128;
  const int* srcArr = ei;
  const int* dstArr = ei + E;

  // ---- workspace carve-up (fp32) ----
  float* W = (float*)d_ws;
  size_t o = 0;
  float* h1     = W + o; o += (size_t)N * 256;  // x@gat1_w, reused for h2
  float* gatout = W + o; o += (size_t)N * 256;  // attention output (both layers)
  float* hsel   = W + o; o += (size_t)N * 64;   // selu(lin) output (both layers)
  float* as_    = W + o; o += (size_t)N * 4;
  float* ad_    = W + o; o += (size_t)N * 4;
  unsigned* mx  = (unsigned*)(W + o); o += (size_t)N * 4;
  float* sm     = W + o; o += (size_t)N * 4;
  float* elog   = W + o; o += (size_t)Et * 4;
  float* psum   = W + o; o += (size_t)G * 64;
  unsigned* pmax= (unsigned*)(W + o); o += (size_t)G * 64;
  float* cnt    = W + o; o += (size_t)G;
  float* cat    = W + o; o += (size_t)G * 256;
  float* z1     = W + o; o += (size_t)G * 128;
  float* z2     = W + o; o += (size_t)G * 64;
  float* z3     = W + o; o += (size_t)G * 64;
  float* z4     = W + o; o += (size_t)G * 64;
  (void)ws_size; (void)n_in; (void)out_size;

  // ---- layer 1 ----
  gemm(x, g1w, nullptr, h1, N, F_in, 256, 0, stream);                 // h1 = x@W1
  run_attention(h1, g1as, g1ad, g1b, srcArr, dstArr,
                as_, ad_, mx, sm, elog, gatout, N, E, Et, stream);    // gatout
  gemm(gatout, l1w, l1b, hsel, N, 256, 64, 1, stream);                // selu(lin1)
  run_pool(hsel, batch, psum, pmax, cnt, cat, 0, N, G, stream);       // x1

  // ---- layer 2 ----
  gemm(hsel, g2w, nullptr, h1, N, 64, 256, 0, stream);                // h2 = hsel@W2
  run_attention(h1, g2as, g2ad, g2b, srcArr, dstArr,
                as_, ad_, mx, sm, elog, gatout, N, E, Et, stream);
  gemm(gatout, l2w, l2b, hsel, N, 256, 64, 1, stream);                // selu(lin2)
  run_pool(hsel, batch, psum, pmax, cnt, cat, 128, N, G, stream);     // x2

  // ---- graph-level head (all WMMA) ----
  gemm(cat, smw, smb, z1, G, 256, 128, 0, stream);  // z = cat@sum_w + sum_b
  gemm(z1, s1w, s1b, z2, G, 128, 64, 1, stream);    // selu
  gemm(z2, s2w, s2b, z3, G, 64, 64, 1, stream);     // selu
  gemm(z3, s3w, s3b, z4, G, 64, 64, 1, stream);     // selu
  gemm(z4, rw, rb, out, G, 64, 32, 0, stream);      // [G,32] -> d_out
}